// MambaLayer_3556232921445
// MI455X (gfx1250) — compile-verified
//
#include <hip/hip_runtime.h>
#include <math.h>

typedef __attribute__((ext_vector_type(2))) float v2f;
typedef __attribute__((ext_vector_type(8))) float v8f;

#define DIM      192
#define D_STATE  16
#define D_INNER  384
#define DT_RANK  12
#define T_LEN    16
#define L_TOK    4096
#define XDBL_PAD 48
#define SCHUNK   64

__device__ __forceinline__ float sigmoidf_(float v) { return 1.0f / (1.0f + __expf(-v)); }
__device__ __forceinline__ float siluf_(float v)    { return v * sigmoidf_(v); }

// ---- gfx1250 async-to-LDS path (guarded; falls back to sync copy) ----------
#if defined(__AMDGCN__) && __has_builtin(__builtin_amdgcn_global_load_async_to_lds_b32) && \
    __has_builtin(__builtin_amdgcn_s_wait_asynccnt)
#define HAVE_ASYNC_LDS 1
#else
#define HAVE_ASYNC_LDS 0
#endif

typedef __attribute__((address_space(1))) int g_int;
typedef __attribute__((address_space(3))) int l_int;

__device__ __forceinline__ void async_cp_f32(float* lds_dst, const float* g_src) {
#if HAVE_ASYNC_LDS
  __builtin_amdgcn_global_load_async_to_lds_b32((g_int*)g_src, (l_int*)lds_dst, 0, 0);
#else
  *lds_dst = *g_src;
#endif
}
__device__ __forceinline__ void async_wait_all() {
#if HAVE_ASYNC_LDS
  __builtin_amdgcn_s_wait_asynccnt(0);
#endif
}

// ---------------------------------------------------------------------------
// K1: 4x4 spatial avg-pool + LayerNorm1.  One block (192 threads) per token.
// ---------------------------------------------------------------------------
__global__ void pool_ln1_kernel(const float* __restrict__ x,
                                const float* __restrict__ g, const float* __restrict__ b,
                                float* __restrict__ tokens, float* __restrict__ y_ln) {
  const int l = blockIdx.x;            // 0..4095
  const int c = threadIdx.x;           // 0..191
  const int t = l >> 8, hs = (l >> 4) & 15, ws = l & 15;
  const float* xp = x + (((size_t)c * T_LEN + t) * 64 + hs * 4) * 64 + ws * 4;
  float s = 0.f;
#pragma unroll
  for (int i = 0; i < 4; ++i)
#pragma unroll
    for (int j = 0; j < 4; ++j) s += xp[i * 64 + j];
  float v = s * (1.0f / 16.0f);

  __shared__ float red[DIM], red2[DIM];
  __shared__ float s_mean, s_rstd;
  red[c] = v; red2[c] = v * v;
  __syncthreads();
  if (c == 0) {
    float su = 0.f, sq = 0.f;
    for (int i = 0; i < DIM; ++i) { su += red[i]; sq += red2[i]; }
    float m = su / (float)DIM;
    float var = sq / (float)DIM - m * m;
    s_mean = m; s_rstd = rsqrtf(var + 1e-5f);
  }
  __syncthreads();
  tokens[(size_t)l * DIM + c] = v;
  y_ln[(size_t)l * DIM + c] = (v - s_mean) * s_rstd * g[c] + b[c];
}

// ---------------------------------------------------------------------------
// Zero-pad W_xp (44x384) -> (48x384) so the GEMM loop has no predication.
// ---------------------------------------------------------------------------
__global__ void pad_wxp_kernel(const float* __restrict__ W_xp, float* __restrict__ Wp) {
  int idx = blockIdx.x * blockDim.x + threadIdx.x;   // 48*384
  if (idx >= XDBL_PAD * D_INNER) return;
  int n = idx / D_INNER;
  Wp[idx] = (n < DT_RANK + 2 * D_STATE) ? W_xp[idx] : 0.f;
}

// ---------------------------------------------------------------------------
// f32 WMMA GEMM:  D[M,N] = A[M,K] * W[N,K]^T    (row-major everywhere)
// One wave owns a 16 x (16*NTW) strip; K advances 4 per V_WMMA_F32_16X16X4_F32.
// Register double-buffering: the k+4 fragments are loaded BEFORE the WMMA
// chain consuming the k fragments, so L2 latency hides behind the matrix ops
// (one loadcnt wait per K-step instead of one per WMMA).  Last iteration
// wraps its prefetch index to 0 -> loop stays branch-free, EXEC all-ones.
// All strides/offsets keep v2f loads 8-byte aligned.
// ---------------------------------------------------------------------------
template <int NTW>
__global__ void gemm_wmma_f32(const float* __restrict__ A, const float* __restrict__ W,
                              float* __restrict__ Dm,
                              int N_groups, int K, int lda, int ldw, int ldd) {
  const int wid   = (blockIdx.x * blockDim.x + threadIdx.x) >> 5;
  const int lane  = threadIdx.x & 31;
  const int mtile = wid / N_groups;
  const int ngrp  = wid % N_groups;
  const int l15   = lane & 15;
  const int koff  = (lane < 16) ? 0 : 2;   // K-pair held by this half-wave
  const float* ap = A + (size_t)(mtile * 16 + l15) * lda + koff;
  const float* wp[NTW];
#pragma unroll
  for (int j = 0; j < NTW; ++j)
    wp[j] = W + (size_t)((ngrp * NTW + j) * 16 + l15) * ldw + koff;

  v8f acc[NTW];
#pragma unroll
  for (int j = 0; j < NTW; ++j) acc[j] = (v8f){0.f,0.f,0.f,0.f,0.f,0.f,0.f,0.f};

  // prologue: fragments for k0 = 0
  v2f a_cur = *(const v2f*)(ap);
  v2f b_cur[NTW];
#pragma unroll
  for (int j = 0; j < NTW; ++j) b_cur[j] = *(const v2f*)(wp[j]);

  for (int k0 = 0; k0 < K; k0 += 4) {
    const int kn = (k0 + 4 < K) ? (k0 + 4) : 0;      // uniform; wraps on last
    v2f a_nxt = *(const v2f*)(ap + kn);
    v2f b_nxt[NTW];
#pragma unroll
    for (int j = 0; j < NTW; ++j) b_nxt[j] = *(const v2f*)(wp[j] + kn);
    __builtin_prefetch(ap + k0 + 64, 0, 3);          // global_prefetch_b8
#pragma unroll
    for (int j = 0; j < NTW; ++j)
      acc[j] = __builtin_amdgcn_wmma_f32_16x16x4_f32(
          /*neg_a=*/false, a_cur, /*neg_b=*/false, b_cur[j],
          /*c_mod=*/(short)0, acc[j], /*reuse_a=*/false, /*reuse_b=*/false);
    a_cur = a_nxt;
#pragma unroll
    for (int j = 0; j < NTW; ++j) b_cur[j] = b_nxt[j];
  }

  const int mbase = mtile * 16 + ((lane < 16) ? 0 : 8);
#pragma unroll
  for (int j = 0; j < NTW; ++j) {
    const int nstore = (ngrp * NTW + j) * 16 + l15;
#pragma unroll
    for (int r = 0; r < 8; ++r)
      Dm[(size_t)(mbase + r) * ldd + nstore] = acc[j][r];
  }
}

// ---------------------------------------------------------------------------
// K3: causal depthwise conv1d (width 4) along L + SiLU.  xc = xz[:, :384].
// ---------------------------------------------------------------------------
__global__ void conv_silu_kernel(const float* __restrict__ xz,
                                 const float* __restrict__ cw, const float* __restrict__ cb,
                                 float* __restrict__ xs) {
  int idx = blockIdx.x * blockDim.x + threadIdx.x;   // l*384 + d
  if (idx >= L_TOK * D_INNER) return;
  int l = idx / D_INNER, d = idx - l * D_INNER;
  float acc = cb[d];
#pragma unroll
  for (int h = 0; h < 4; ++h) {
    int ls = l + h - 3;
    float xv = (ls >= 0) ? xz[(size_t)ls * 768 + d] : 0.f;
    acc += cw[d * 4 + h] * xv;
  }
  xs[idx] = siluf_(acc);
}

// dt = softplus(dt_pre + b_dt), in-place
__global__ void dt_softplus_kernel(float* __restrict__ dt, const float* __restrict__ b_dt) {
  int idx = blockIdx.x * blockDim.x + threadIdx.x;
  if (idx >= L_TOK * D_INNER) return;
  int d = idx % D_INNER;
  float v = dt[idx] + b_dt[d];
  dt[idx] = (v > 20.f) ? v : log1pf(__expf(v));
}

// ---------------------------------------------------------------------------
// Selective scan.  Thread = (d_local, s); 16 channels x 16 states / block.
// h_l = exp(dt*A)*h_{l-1} + dt*B_l*x_l ;  y_l[d] = sum_s h[d,s]*C_l[s]
// 64-step chunks of dt/xs/B/C are double-buffered in LDS via the gfx1250
// async-to-LDS path (ASYNCcnt), overlapping memory with the serial recurrence.
// Reduction over s via wave32 shuffles (width 16); gating fused on writeout.
// ---------------------------------------------------------------------------
__global__ void scan_kernel(const float* __restrict__ dt, const float* __restrict__ xs,
                            const float* __restrict__ xdbl, const float* __restrict__ A_log,
                            const float* __restrict__ Dv, const float* __restrict__ xz,
                            float* __restrict__ y_act) {
  __shared__ float sBC[2][SCHUNK][32];   // B(16)|C(16) per token   16 KB
  __shared__ float sDT[2][SCHUNK][16];   //                          8 KB
  __shared__ float sXS[2][SCHUNK][16];   //                          8 KB
  const int tid = threadIdx.x;
  const int dl  = tid >> 4;              // 0..15
  const int s   = tid & 15;              // 0..15
  const int d   = blockIdx.x * 16 + dl;  // 0..383
  const int d0  = blockIdx.x * 16;
  const float a  = -__expf(A_log[d * D_STATE + s]);
  const float Dd = Dv[d];

  auto stage = [&](int c, int bi) {
    const int l0 = c * SCHUNK;
#pragma unroll
    for (int i = 0; i < 8; ++i) {              // 64*32 = 2048 elems
      int e = tid + i * 256;
      int ll = e >> 5, col = e & 31;
      async_cp_f32(&sBC[bi][ll][col],
                   &xdbl[(size_t)(l0 + ll) * XDBL_PAD + DT_RANK + col]);
    }
#pragma unroll
    for (int i = 0; i < 4; ++i) {              // 64*16 = 1024 elems each
      int e = tid + i * 256;
      int ll = e >> 4, col = e & 15;
      async_cp_f32(&sDT[bi][ll][col], &dt[(size_t)(l0 + ll) * D_INNER + d0 + col]);
      async_cp_f32(&sXS[bi][ll][col], &xs[(size_t)(l0 + ll) * D_INNER + d0 + col]);
    }
  };

  stage(0, 0);
  float h = 0.f;
  int cur = 0;
  const int NCH = L_TOK / SCHUNK;
  for (int c = 0; c < NCH; ++c) {
    async_wait_all();
    __syncthreads();
    if (c + 1 < NCH) stage(c + 1, cur ^ 1);
    const int l0 = c * SCHUNK;
    for (int ll = 0; ll < SCHUNK; ++ll) {
      float dtl = sDT[cur][ll][dl];
      float xsl = sXS[cur][ll][dl];
      float Bl  = sBC[cur][ll][s];
      float Cl  = sBC[cur][ll][16 + s];
      h = __expf(dtl * a) * h + dtl * Bl * xsl;
      float p = h * Cl;
      p += __shfl_xor(p, 1, 16);
      p += __shfl_xor(p, 2, 16);
      p += __shfl_xor(p, 4, 16);
      p += __shfl_xor(p, 8, 16);
      if (s == 0) {
        int l = l0 + ll;
        float zv = xz[(size_t)l * 768 + D_INNER + d];
        y_act[(size_t)l * D_INNER + d] = (p + xsl * Dd) * siluf_(zv);
      }
    }
    cur ^= 1;
  }
}

// ---------------------------------------------------------------------------
// Residual add + LayerNorm2
// ---------------------------------------------------------------------------
__global__ void add_ln2_kernel(const float* __restrict__ tokens, const float* __restrict__ ym,
                               const float* __restrict__ g, const float* __restrict__ b,
                               float* __restrict__ out_ln) {
  const int l = blockIdx.x;
  const int c = threadIdx.x;
  float v = tokens[(size_t)l * DIM + c] + ym[(size_t)l * DIM + c];
  __shared__ float red[DIM], red2[DIM];
  __shared__ float s_mean, s_rstd;
  red[c] = v; red2[c] = v * v;
  __syncthreads();
  if (c == 0) {
    float su = 0.f, sq = 0.f;
    for (int i = 0; i < DIM; ++i) { su += red[i]; sq += red2[i]; }
    float m = su / (float)DIM;
    float var = sq / (float)DIM - m * m;
    s_mean = m; s_rstd = rsqrtf(var + 1e-5f);
  }
  __syncthreads();
  out_ln[(size_t)l * DIM + c] = (v - s_mean) * s_rstd * g[c] + b[c];
}

// ---------------------------------------------------------------------------
// Per-channel spatial pool per t, 16-pt DFT mags (bins 1..7),
// sigmoid of mean -> weight[c].  One block per channel.
// ---------------------------------------------------------------------------
__global__ void fft_weight_kernel(const float* __restrict__ out_ln, float* __restrict__ weight) {
  const int c   = blockIdx.x;    // 0..191
  const int tid = threadIdx.x;   // 0..255
  __shared__ float red[256];
  __shared__ float pooled[T_LEN];
  for (int t = 0; t < T_LEN; ++t) {
    red[tid] = out_ln[(size_t)(t * 256 + tid) * DIM + c];
    __syncthreads();
    for (int st = 128; st > 0; st >>= 1) {
      if (tid < st) red[tid] += red[tid + st];
      __syncthreads();
    }
    if (tid == 0) pooled[t] = red[0] * (1.0f / 256.0f);
    __syncthreads();
  }
  if (tid == 0) {
    float msum = 0.f;
    for (int k = 1; k <= 7; ++k) {
      float re = 0.f, im = 0.f;
      for (int t = 0; t < T_LEN; ++t) {
        float ang = -2.0f * 3.14159265358979323846f * (float)(k * t) / (float)T_LEN;
        re += pooled[t] * __cosf(ang);
        im += pooled[t] * __sinf(ang);
      }
      msum += sqrtf(re * re + im * im);
    }
    weight[c] = sigmoidf_(msum / 7.0f);
  }
}

// ---------------------------------------------------------------------------
// Bilinear 16->64 upsample in H,W (half-pixel, edge-clamp; T identity),
// scale by weight[c], gate by sigmoid(x).
// ---------------------------------------------------------------------------
__global__ void upsample_gate_kernel(const float* __restrict__ out_ln,
                                     const float* __restrict__ weight,
                                     const float* __restrict__ x,
                                     float* __restrict__ out) {
  size_t idx = (size_t)blockIdx.x * blockDim.x + threadIdx.x;  // (c,t,h,w)
  int w = (int)(idx & 63);
  int h = (int)((idx >> 6) & 63);
  int t = (int)((idx >> 12) & 15);
  int c = (int)(idx >> 16);
  float fh = (h + 0.5f) * 0.25f - 0.5f;
  float fw = (w + 0.5f) * 0.25f - 0.5f;
  int i0 = (int)floorf(fh); float dh = fh - (float)i0;
  int j0 = (int)floorf(fw); float dw = fw - (float)j0;
  int i1 = i0 + 1, j1 = j0 + 1;
  i0 = min(max(i0, 0), 15); i1 = min(max(i1, 0), 15);
  j0 = min(max(j0, 0), 15); j1 = min(max(j1, 0), 15);
  const size_t tb = (size_t)(t << 8);
  float v00 = out_ln[(tb + (i0 << 4) + j0) * DIM + c];
  float v01 = out_ln[(tb + (i0 << 4) + j1) * DIM + c];
  float v10 = out_ln[(tb + (i1 << 4) + j0) * DIM + c];
  float v11 = out_ln[(tb + (i1 << 4) + j1) * DIM + c];
  float v = (v00 * (1.f - dw) + v01 * dw) * (1.f - dh) +
            (v10 * (1.f - dw) + v11 * dw) * dh;
  v *= weight[c];
  out[idx] = v * sigmoidf_(x[idx]);
}

// ---------------------------------------------------------------------------
extern "C" void kernel_launch(void* const* d_in, const int* in_sizes, int n_in,
                              void* d_out, int out_size, void* d_ws, size_t ws_size,
                              hipStream_t stream) {
  const float* x      = (const float*)d_in[0];
  const float* ln1_g  = (const float*)d_in[1];
  const float* ln1_b  = (const float*)d_in[2];
  const float* ln2_g  = (const float*)d_in[3];
  const float* ln2_b  = (const float*)d_in[4];
  const float* W_in   = (const float*)d_in[5];
  const float* conv_w = (const float*)d_in[6];
  const float* conv_b = (const float*)d_in[7];
  const float* W_xp   = (const float*)d_in[8];
  const float* W_dt   = (const float*)d_in[9];
  const float* b_dt   = (const float*)d_in[10];
  const float* A_log  = (const float*)d_in[11];
  const float* Dv     = (const float*)d_in[12];
  const float* W_out  = (const float*)d_in[13];
  float* out = (float*)d_out;
  float* ws  = (float*)d_ws;

  float* tokens  = ws;                         // 4096*192
  float* y_ln    = tokens  + L_TOK * DIM;      // 4096*192
  float* xz      = y_ln    + L_TOK * DIM;      // 4096*768
  float* xs      = xz      + L_TOK * 768;      // 4096*384
  float* x_dbl   = xs      + L_TOK * D_INNER;  // 4096*48
  float* dtbuf   = x_dbl   + L_TOK * XDBL_PAD; // 4096*384
  float* y_act   = dtbuf   + L_TOK * D_INNER;  // 4096*384
  float* out_m   = y_act   + L_TOK * D_INNER;  // 4096*192
  float* out_ln  = out_m   + L_TOK * DIM;      // 4096*192
  float* weight  = out_ln  + L_TOK * DIM;      // 192
  float* wxp_pad = weight  + DIM;              // 48*384

  pad_wxp_kernel<<<(XDBL_PAD * D_INNER + 255) / 256, 256, 0, stream>>>(W_xp, wxp_pad);

  pool_ln1_kernel<<<L_TOK, DIM, 0, stream>>>(x, ln1_g, ln1_b, tokens, y_ln);

  // xz = y_ln(4096x192) * W_in(768x192)^T : 256 Mtiles x 12 groups of 4 Ntiles
  gemm_wmma_f32<4><<<384, 256, 0, stream>>>(y_ln, W_in, xz, 12, 192, 192, 192, 768);

  conv_silu_kernel<<<(L_TOK * D_INNER) / 256, 256, 0, stream>>>(xz, conv_w, conv_b, xs);

  // x_dbl = xs(4096x384) * wxp_pad(48x384)^T : one group of 3 Ntiles
  gemm_wmma_f32<3><<<32, 256, 0, stream>>>(xs, wxp_pad, x_dbl, 1, 384, 384, 384, XDBL_PAD);

  // dt_pre = x_dbl[:, :12] * W_dt(384x12)^T : K=12 (3 WMMA steps), 6 groups of 4
  gemm_wmma_f32<4><<<192, 256, 0, stream>>>(x_dbl, W_dt, dtbuf, 6, 12, XDBL_PAD, 12, D_INNER);

  dt_softplus_kernel<<<(L_TOK * D_INNER) / 256, 256, 0, stream>>>(dtbuf, b_dt);

  scan_kernel<<<D_INNER / 16, 256, 0, stream>>>(dtbuf, xs, x_dbl, A_log, Dv, xz, y_act);

  // out_m = y_act(4096x384) * W_out(192x384)^T : 3 groups of 4 Ntiles
  gemm_wmma_f32<4><<<96, 256, 0, stream>>>(y_act, W_out, out_m, 3, 384, 384, 384, DIM);

  add_ln2_kernel<<<L_TOK, DIM, 0, stream>>>(tokens, out_m, ln2_g, ln2_b, out_ln);

  fft_weight_kernel<<<DIM, 256, 0, stream>>>(out_ln, weight);

  upsample_gate_kernel<<<(192 * 16 * 64 * 64) / 256, 256, 0, stream>>>(out_ln, weight, x, out);
}